// LFQ_56453050139323
// MI455X (gfx1250) — compile-verified
//
#include <hip/hip_runtime.h>
#include <hip/hip_bf16.h>

// Problem constants (from reference): x is (4, 14, 32, 32) f32.
#define DDIM      14
#define KCODES    16384
#define NSAMP     4096          // 4 * 32 * 32
#define NTILES    (KCODES / 16) // 1024 code tiles of 16
#define WAVES     8
#define TILES_PW  (NTILES / WAVES) // 128 tiles per wave
#define QELEMS    (NSAMP * DDIM)   // 57344
#define TEMP_INV  100.0f           // 1/TEMPERATURE
#define EPSF      1e-9f

typedef float v2f __attribute__((ext_vector_type(2)));
typedef float v8f __attribute__((ext_vector_type(8)));

// ---------------------------------------------------------------------------
// One 16(samples) x 16(codes) logit-accumulator tile via 4x V_WMMA_F32_16X16X4_F32.
// A (16x16, K padded from 14 with zeros) is preloaded in a[4] per the ISA f32
// A-layout: lane -> row M = lane&15, VGPR pair holds K = kb*4 + 2*(lane>>4) (+1).
// B (16x16 of +-1) is synthesized from the code's bits: B[j][c] = bit_j(code)?+1:-1
// (little-endian bit order, matching the reference codebook). Bits 14/15 multiply
// zero-padded A columns, so their value is irrelevant.
// ---------------------------------------------------------------------------
__device__ __forceinline__ v8f lfq_tile_acc(const v2f a[4], int ktile, int lane) {
  const int c    = lane & 15;
  const int half = lane >> 4;
  const int code = ktile * 16 + c;
  v8f acc = {0.f, 0.f, 0.f, 0.f, 0.f, 0.f, 0.f, 0.f};
#pragma unroll
  for (int kb = 0; kb < 4; ++kb) {
    const int j0 = kb * 4 + 2 * half;
    v2f bv;
    bv.x = ((code >> j0) & 1)       ? 1.0f : -1.0f;
    bv.y = ((code >> (j0 + 1)) & 1) ? 1.0f : -1.0f;
    acc = __builtin_amdgcn_wmma_f32_16x16x4_f32(false, a[kb], false, bv,
                                                (short)0, acc, false, false);
  }
  return acc;
}

// ---------------------------------------------------------------------------
// Kernel 1: per-sample prep. Computes q = sign(x), big-endian indices,
// normalized xn written as zero-padded 16-wide rows (WMMA A staging), and
// zeroes the avg_probs / entropy accumulators (ws is poisoned each run).
// ---------------------------------------------------------------------------
__global__ __launch_bounds__(256) void lfq_prep_kernel(
    const float* __restrict__ x, float* __restrict__ q_out,
    float* __restrict__ idx_out, float* __restrict__ ws_xn,
    float* __restrict__ ws_avg, float* __restrict__ ws_entro) {
  const int n = blockIdx.x * 256 + threadIdx.x;
  if (n < NSAMP) {
    const int b  = n >> 10;        // 1024 samples per batch image
    const int hw = n & 1023;
    const float* xb = x + (size_t)b * DDIM * 1024 + hw;
    float* qb = q_out + (size_t)b * DDIM * 1024 + hw;

    float v[DDIM];
    float ss = 0.f;
#pragma unroll
    for (int j = 0; j < DDIM; ++j) {
      v[j] = xb[j * 1024];
      ss += v[j] * v[j];
    }
    const float inv = rsqrtf(ss);

    int idx = 0;
#pragma unroll
    for (int j = 0; j < DDIM; ++j) {
      const bool pos = v[j] > 0.0f;
      qb[j * 1024] = pos ? 1.0f : -1.0f;
      if (pos) idx |= (1 << (DDIM - 1 - j)); // big-endian index mask
    }
    idx_out[n] = (float)idx;

    float xn[16];
#pragma unroll
    for (int j = 0; j < DDIM; ++j) xn[j] = v[j] * inv;
    xn[14] = 0.f;
    xn[15] = 0.f;
    float4* dst = (float4*)(ws_xn + (size_t)n * 16);
#pragma unroll
    for (int i = 0; i < 4; ++i)
      dst[i] = make_float4(xn[4 * i], xn[4 * i + 1], xn[4 * i + 2], xn[4 * i + 3]);

    // zero avg_probs (4096 threads * 4 = 16384)
    ((float4*)ws_avg)[n] = make_float4(0.f, 0.f, 0.f, 0.f);
  }
  if (n == 0) *ws_entro = 0.f;
}

// ---------------------------------------------------------------------------
// Kernel 2: fused entropy. One block = 16 samples; 8 waves each sweep 128
// code tiles. Pass 1: WMMA + online softmax (m, s=sum e^{z-m}, t=sum e^{z-m}z)
// -> per-sample entropy H = m + log s - t/s, atomically accumulated.
// Pass 2: WMMA recompute, p = e^{z-M}/S accumulated into global avg_probs.
// D-tile layout: VGPR i of lane L holds row M = i + 8*(L>>4), col N = L&15.
// ---------------------------------------------------------------------------
__global__ __launch_bounds__(256) void lfq_entropy_kernel(
    const float* __restrict__ ws_xn, float* __restrict__ ws_avg,
    float* __restrict__ ws_entro) {
  const int tid   = threadIdx.x;
  const int lane  = tid & 31;
  const int wave  = tid >> 5;
  const int mtile = blockIdx.x;
  const int row   = lane & 15;
  const int half  = lane >> 4;

  // A fragment: 16 samples x 16 dims, reused for all 2*128 tiles of this wave.
  v2f a[4];
  const float* abase = ws_xn + ((size_t)mtile * 16 + row) * 16 + 2 * half;
#pragma unroll
  for (int kb = 0; kb < 4; ++kb) a[kb] = *(const v2f*)(abase + kb * 4);

  // ---------------- pass 1: online softmax statistics ----------------
  float m[8], s[8], t[8];
#pragma unroll
  for (int i = 0; i < 8; ++i) { m[i] = -3.0e38f; s[i] = 0.f; t[i] = 0.f; }

  const int kt0 = wave * TILES_PW;
  for (int it = 0; it < TILES_PW; ++it) {
    const v8f acc = lfq_tile_acc(a, kt0 + it, lane);
#pragma unroll
    for (int i = 0; i < 8; ++i) {
      const float z  = (2.0f * TEMP_INV) * acc[i]; // z = 2*dot / T
      const float nm = fmaxf(m[i], z);
      const float e1 = __expf(m[i] - nm);
      const float e2 = __expf(z - nm);
      s[i] = s[i] * e1 + e2;
      t[i] = t[i] * e1 + e2 * z;
      m[i] = nm;
    }
  }
  // merge across the 16 lanes holding the same row group
#pragma unroll
  for (int mask = 8; mask >= 1; mask >>= 1) {
#pragma unroll
    for (int i = 0; i < 8; ++i) {
      const float om = __shfl_xor(m[i], mask, 32);
      const float os = __shfl_xor(s[i], mask, 32);
      const float ot = __shfl_xor(t[i], mask, 32);
      const float nm = fmaxf(m[i], om);
      const float e1 = __expf(m[i] - nm);
      const float e2 = __expf(om - nm);
      s[i] = s[i] * e1 + os * e2;
      t[i] = t[i] * e1 + ot * e2;
      m[i] = nm;
    }
  }

  __shared__ float lm[WAVES][16], ls[WAVES][16], lt[WAVES][16];
  __shared__ float fM[16], fIS[16];
  if ((lane & 15) == 0) { // lane 0 -> rows 0..7, lane 16 -> rows 8..15
    const int rb = half * 8;
#pragma unroll
    for (int i = 0; i < 8; ++i) {
      lm[wave][rb + i] = m[i];
      ls[wave][rb + i] = s[i];
      lt[wave][rb + i] = t[i];
    }
  }
  __syncthreads();

  if (tid < 16) { // cross-wave merge + entropy (no WMMA under partial EXEC)
    float M = -3.0e38f, S = 0.f, T = 0.f;
#pragma unroll
    for (int w = 0; w < WAVES; ++w) {
      const float om = lm[w][tid], os = ls[w][tid], ot = lt[w][tid];
      const float nm = fmaxf(M, om);
      const float e1 = __expf(M - nm);
      const float e2 = __expf(om - nm);
      S = S * e1 + os * e2;
      T = T * e1 + ot * e2;
      M = nm;
    }
    float H = M + __logf(S) - T / S; // -sum p log p
    fM[tid]  = M;
    fIS[tid] = 1.0f / S;
#pragma unroll
    for (int mask = 8; mask >= 1; mask >>= 1) H += __shfl_xor(H, mask, 32);
    if (tid == 0) atomicAdd(ws_entro, H);
  }
  __syncthreads();

  // ---------------- pass 2: avg_probs accumulation ----------------
  float Mr[8], Ir[8];
#pragma unroll
  for (int i = 0; i < 8; ++i) {
    Mr[i] = fM[half * 8 + i];
    Ir[i] = fIS[half * 8 + i];
  }
  for (int it = 0; it < TILES_PW; ++it) {
    const int kt = kt0 + it;
    const v8f acc = lfq_tile_acc(a, kt, lane);
    float ps = 0.f;
#pragma unroll
    for (int i = 0; i < 8; ++i)
      ps += __expf((2.0f * TEMP_INV) * acc[i] - Mr[i]) * Ir[i];
    ps += __shfl_xor(ps, 16, 32); // fold row halves: same code, rows 0..7 + 8..15
    if (lane < 16) atomicAdd(&ws_avg[kt * 16 + lane], ps);
  }
}

// ---------------------------------------------------------------------------
// Kernel 3: el = entro_mean - (-sum a log(a+eps)), a = avg_probs/NSAMP.
// ---------------------------------------------------------------------------
__global__ __launch_bounds__(256) void lfq_finalize_kernel(
    const float* __restrict__ ws_avg, const float* __restrict__ ws_entro,
    float* __restrict__ out_el) {
  __shared__ float red[256];
  float partial = 0.f;
  for (int k = threadIdx.x; k < KCODES; k += 256) {
    const float av = ws_avg[k] * (1.0f / (float)NSAMP);
    partial += av * __logf(av + EPSF);
  }
  red[threadIdx.x] = partial;
  __syncthreads();
  for (int stride = 128; stride > 0; stride >>= 1) {
    if (threadIdx.x < stride) red[threadIdx.x] += red[threadIdx.x + stride];
    __syncthreads();
  }
  if (threadIdx.x == 0) {
    const float mean_entro = -red[0];
    const float entro_mean = ws_entro[0] * (1.0f / (float)NSAMP);
    out_el[0] = entro_mean - mean_entro;
  }
}

extern "C" void kernel_launch(void* const* d_in, const int* in_sizes, int n_in,
                              void* d_out, int out_size, void* d_ws, size_t ws_size,
                              hipStream_t stream) {
  (void)in_sizes; (void)n_in; (void)out_size; (void)ws_size;
  const float* x = (const float*)d_in[0];

  // Outputs concatenated flat in return order: q[57344], el[1], indices[4096].
  float* out     = (float*)d_out;
  float* q_out   = out;
  float* el_out  = out + QELEMS;
  float* idx_out = out + QELEMS + 1;

  // Workspace: xn A-tiles (4096*16 f32) | avg_probs (16384 f32) | entropy (1 f32)
  float* ws_xn    = (float*)d_ws;
  float* ws_avg   = ws_xn + (size_t)NSAMP * 16;
  float* ws_entro = ws_avg + KCODES;

  lfq_prep_kernel<<<NSAMP / 256, 256, 0, stream>>>(x, q_out, idx_out, ws_xn,
                                                   ws_avg, ws_entro);
  lfq_entropy_kernel<<<NSAMP / 16, 256, 0, stream>>>(ws_xn, ws_avg, ws_entro);
  lfq_finalize_kernel<<<1, 256, 0, stream>>>(ws_avg, ws_entro, el_out);
}